// CubeSimulator_2997887172807
// MI455X (gfx1250) — compile-verified
//
#include <hip/hip_runtime.h>

typedef float v2f __attribute__((ext_vector_type(2)));
typedef float v8f __attribute__((ext_vector_type(8)));

#define OUT_RES   64

__device__ __forceinline__ float fexp2(float x) { return __builtin_amdgcn_exp2f(x); }

__launch_bounds__(256)
__global__ void cube_sim_kernel(const float* __restrict__ freqs,
                                const float* __restrict__ p_inc,
                                const float* __restrict__ p_rot,
                                const float* __restrict__ p_lb,
                                const float* __restrict__ p_vs,
                                const float* __restrict__ p_vmax,
                                const float* __restrict__ p_rturn,
                                const float* __restrict__ p_i0,
                                const float* __restrict__ p_rd,
                                const float* __restrict__ p_hz,
                                float* __restrict__ out)
{
    // Per block: 8 waves = 2 patches (4x4 input pixels each) x 4 z-slices.
    __shared__ float red[2][4][256];

    const int tid   = threadIdx.x;
    const int L     = tid & 31;          // lane in wave32
    const int wb    = tid >> 5;          // wave in block (0..7)
    const int pb    = wb >> 2;           // patch in block (0..1)
    const int zs    = wb & 3;            // z-slice (0..3)
    const int laneN = L & 15;            // WMMA column / pixel-in-patch
    const int kh    = (L >> 4) << 1;     // K half offset: 0 or 2

    const int gp = blockIdx.x * 2 + pb;  // global patch id, 0..1023 (32x32 patches)
    const int bx = (gp >> 5) << 2;       // patch base in x (0..124)
    const int by = (gp & 31) << 2;       // patch base in y
    const int xl = laneN >> 2, yl = laneN & 3;
    const int ix = bx + xl, iy = by + yl;

    // Scalar parameters
    const float inc   = p_inc[0],  rot  = p_rot[0];
    const float sig   = p_lb[0],   vsh  = p_vs[0];
    const float vmax  = p_vmax[0], rtrn = p_rturn[0];
    const float i0    = p_i0[0],   rd   = p_rd[0], hz = p_hz[0];
    const float f0    = freqs[0],  f15  = freqs[15];

    const float C_KMS  = 299792.458f;
    const float F_REST = 230.538f;
    const float L2E    = 1.4426950408889634f;
    const float TWO_PI = 6.2831853071795864f;

    const float ci = cosf(inc), si = sinf(inc);
    const float cr = cosf(rot), sr = sinf(rot);

    // R = Rx*Rz => rx = cr*X - sr*Y (z-invariant!); ry = ci*(sr*X+cr*Y) - si*Z;
    // rz = si*(sr*X+cr*Y) + ci*Z;  v_los = -si * v_abs * rx / rad.
    const float step = 2000.0f / 127.0f;
    const float X  = -1000.0f + step * (float)ix;
    const float Y  = -1000.0f + step * (float)iy;
    const float rx  = cr * X - sr * Y;
    const float s1  = sr * X + cr * Y;
    const float ryc = ci * s1;
    const float rzc = si * s1;
    const float msi = -si * rx;          // per-lane constant

    // Velocity labels: vl(fi) = vlA + vlB*fi
    const float df  = (f15 - f0) * (1.0f / 63.0f);
    const float g   = C_KMS / F_REST;
    const float vlA = g * (F_REST - f0 + 2.0f * df);
    const float vlB = -g * df;
    const float bias_kh = vlB * (float)kh;

    const float inv_sig  = 1.0f / (sig * sig);
    const float mq       = -inv_sig * L2E;          // exp(-d^2/s^2) = exp2(d^2*mq)
    const float inv_rtrn = 1.0f / rtrn;
    const float vmax2pi  = vmax * 0.63661977236758134f; // 2/pi
    const float inv_rd   = 1.0f / rd;
    const float inv_hz   = 1.0f / hz;
    const float log2_i0  = log2f(i0);               // fold i0 into the exponent

    const v2f mqv  = { mq, mq };
    const v2f dstp = { 4.0f * vlB, 4.0f * vlB };

    v8f acc = {0.f,0.f,0.f,0.f,0.f,0.f,0.f,0.f};

    const int z0 = zs * 32;
    for (int iz = z0; iz < z0 + 32; ++iz) {
        const float Z   = -1000.0f + step * (float)iz;
        const float ry  = ryc - si * Z;
        const float rz  = rzc + ci * Z;
        const float rad = sqrtf(rx * rx + ry * ry + 1e-12f);
        const float vab = vmax2pi * atanf(rad * inv_rtrn);
        const float vlos = msi * vab * __builtin_amdgcn_rcpf(rad);
        const float u   = vlos + vsh;
        // log2 of intensity: I = i0 * exp2(t)  ->  fold into kde exponent
        const float logI = log2_i0 - (rad * inv_rd + fabsf(rz) * inv_hz) * L2E;
        const v2f logIv = { logI, logI };

        v2f d;                              // diffs at fi = kh, kh+1 (chunk c=0)
        d.x = (vlA - u) + bias_kh;
        d.y = d.x + vlB;
        #pragma unroll
        for (int c = 0; c < 16; ++c) {
            const v2f sq  = d * d;          // v_pk_mul_f32
            const v2f arg = sq * mqv + logIv; // v_pk_fma_f32
            v2f b;
            b.x = fexp2(arg.x);             // kde * intensity, fused in exp2
            b.y = fexp2(arg.y);
            const float a = (laneN == c) ? 0.25f : 0.0f;  // freq-pool selector row
            v2f av; av.x = a; av.y = a;
            // D[fo, pixel] += A(16x4) * B(4x16): accumulate + freq pooling on matrix pipe
            acc = __builtin_amdgcn_wmma_f32_16x16x4_f32(
                false, av, false, b, (short)0, acc, false, false);
            d = d + dstp;                   // v_pk_add_f32: advance 4 freqs
        }
    }

    // Combine the 4 z-slice partials through LDS.
    #pragma unroll
    for (int v = 0; v < 8; ++v) red[pb][zs][L * 8 + v] = acc[v];
    __syncthreads();

    if (zs == 0) {
        const float norm = 0.25f / sqrtf(TWO_PI * sig * sig); // KDE norm * spatial mean
        const int fo_base = (L >> 4) << 3;                    // C/D: M = v + 8*(lane/16)
        #pragma unroll
        for (int v = 0; v < 8; ++v) {
            float t = red[pb][0][L * 8 + v] + red[pb][1][L * 8 + v]
                    + red[pb][2][L * 8 + v] + red[pb][3][L * 8 + v];
            // 2x2 spatial pool: patch col n = 4*xl+yl; quad = {n, n^1, n^4, n^5}
            t += __shfl_xor(t, 1, 32);
            t += __shfl_xor(t, 4, 32);
            if ((laneN & 5) == 0) {
                const int oxl = laneN >> 3;
                const int oyl = (laneN >> 1) & 1;
                const int ox  = (bx >> 1) + oxl;
                const int oy  = (by >> 1) + oyl;
                const int fo  = fo_base + v;
                out[fo * (OUT_RES * OUT_RES) + ox * OUT_RES + oy] = t * norm;
            }
        }
    }
}

extern "C" void kernel_launch(void* const* d_in, const int* in_sizes, int n_in,
                              void* d_out, int out_size, void* d_ws, size_t ws_size,
                              hipStream_t stream) {
    (void)in_sizes; (void)n_in; (void)d_ws; (void)ws_size; (void)out_size;
    const float* freqs  = (const float*)d_in[0];
    const float* inc    = (const float*)d_in[1];
    const float* rot    = (const float*)d_in[2];
    const float* lb     = (const float*)d_in[3];
    const float* vs     = (const float*)d_in[4];
    const float* vmax   = (const float*)d_in[5];
    const float* rturn  = (const float*)d_in[6];
    const float* i0     = (const float*)d_in[7];
    const float* rd     = (const float*)d_in[8];
    const float* hz     = (const float*)d_in[9];
    float* out = (float*)d_out;

    // 1024 patches (32x32 of 4x4 pixels) x 4 z-slices = 4096 waves; 8 waves/block.
    cube_sim_kernel<<<512, 256, 0, stream>>>(freqs, inc, rot, lb, vs, vmax,
                                             rturn, i0, rd, hz, out);
}